// TopkAttnFunction_Pytorch_32143535243417
// MI455X (gfx1250) — compile-verified
//
#include <hip/hip_runtime.h>

typedef float v2f __attribute__((ext_vector_type(2)));
typedef float v8f __attribute__((ext_vector_type(8)));

// Problem constants from the reference: B=2, Lq=4096, Lk=4096, H=8, C=64, K=32
static constexpr int H_   = 8;
static constexpr int C_   = 64;
static constexpr int K_   = 32;
static constexpr int LK_  = 4096;
static constexpr int HC_  = H_ * C_;      // 512: stride (in floats) between rows of key/value
static constexpr int LQH_SHIFT = 15;      // Lq*H = 32768 = 1<<15 (queries per batch)
static constexpr int WAVES = 8;           // waves per block (256 threads = 8 x wave32)
static constexpr float SCALE_INV = 0.125f; // 1/SCALE, SCALE=8

__global__ __launch_bounds__(WAVES * 32)
void topk_attn_wmma_kernel(const float* __restrict__ query,
                           const float* __restrict__ key,
                           const float* __restrict__ value,
                           const long long* __restrict__ pos,
                           float* __restrict__ out,
                           int Q) // total queries = B*Lq*H
{
    __shared__ float  qs[WAVES][C_];
    __shared__ float2 ar[WAVES][K_];   // per key k: {attn_k, bitcast(row_k * HC_)}

    const int w    = threadIdx.x >> 5;
    const int lane = threadIdx.x & 31;
    const int qid  = blockIdx.x * WAVES + w;     // wave-uniform
    if (qid >= Q) return;                        // uniform exit: EXEC stays all-1s inside

    const int h = qid & (H_ - 1);
    const int b = qid >> LQH_SHIFT;
    // Base (in floats) of (b, h) plane inside key/value: ((b*Lk + row)*H + h)*C
    const size_t kvbase = (size_t)b * (size_t)LK_ * HC_ + (size_t)h * C_;

    // ---- Stage q[64] into this wave's LDS slice (broadcast source) ----
    {
        const float2 q2 = reinterpret_cast<const float2*>(query + (size_t)qid * C_)[lane];
        qs[w][2 * lane]     = q2.x;
        qs[w][2 * lane + 1] = q2.y;
    }
    __builtin_amdgcn_wave_barrier(); // scheduling fence: DS store before DS loads (same wave, in-order)

    // ---- Gather: lane k owns key/value row k ----
    const int row = (int)pos[(size_t)qid * K_ + lane];      // 0..Lk-1
    const float* krow = key + kvbase + (size_t)row * HC_;

    float acc = 0.f;
    #pragma unroll
    for (int c4 = 0; c4 < C_ / 4; ++c4) {
        const float4 kv = reinterpret_cast<const float4*>(krow)[c4];       // 256B/lane contiguous
        const float4 qv = reinterpret_cast<const float4*>(&qs[w][0])[c4];  // LDS broadcast
        acc = fmaf(kv.x, qv.x, acc);
        acc = fmaf(kv.y, qv.y, acc);
        acc = fmaf(kv.z, qv.z, acc);
        acc = fmaf(kv.w, qv.w, acc);
    }
    const float s = acc * SCALE_INV;

    // ---- Softmax over the 32 lanes (wave32 shuffle reductions) ----
    float mx = s;
    #pragma unroll
    for (int off = 16; off > 0; off >>= 1)
        mx = fmaxf(mx, __shfl_xor(mx, off, 32));
    const float e = __expf(s - mx);
    float sum = e;
    #pragma unroll
    for (int off = 16; off > 0; off >>= 1)
        sum += __shfl_xor(sum, off, 32);
    const float attn = e / sum;                  // lane k holds attn_k

    // ---- Stage {attn_k, row_offset_k} pairs in LDS; one b128 read rebuilds each k-chunk ----
    {
        float2 p;
        p.x = attn;
        p.y = __int_as_float(row * HC_);
        ar[w][lane] = p;                         // ds_store_b64
    }
    __builtin_amdgcn_wave_barrier();

    // A (16x4 f32): VGPR0 = {K0 | K2}, VGPR1 = {K1 | K3}; only row M=0 nonzero (lanes 0 and 16).
    // B (4x16 f32): VGPR0 = {row K0 (N=lane) | row K2}, VGPR1 = {row K1 | row K3}.
    const int col  = lane & 15;
    const int hsel = (lane >> 4) << 1;           // 0 for lanes 0-15, 2 for lanes 16-31

    const float* vbase = value + kvbase;

    float a0[8], a1[8];
    const float* p0[8];                          // per-lane V row pointers (col folded in)
    const float* p1[8];
    #pragma unroll
    for (int kc = 0; kc < 8; ++kc) {
        const int i0 = 4 * kc + hsel;            // even -> 16B-aligned b128 LDS read
        const float4 f = *reinterpret_cast<const float4*>(&ar[w][i0]);
        // f = {attn[i0], off[i0], attn[i0+1], off[i0+1]}
        a0[kc] = (col == 0) ? f.x : 0.f;         // lane0: attn[4kc],   lane16: attn[4kc+2]
        a1[kc] = (col == 0) ? f.z : 0.f;         // lane0: attn[4kc+1], lane16: attn[4kc+3]
        p0[kc] = vbase + __float_as_int(f.y) + col;
        p1[kc] = vbase + __float_as_int(f.w) + col;
    }

    // ---- out(1x64) = attn(1x32) x V(32x64) via v_wmma_f32_16x16x4_f32 ----
    float r[4];
    #pragma unroll
    for (int ct = 0; ct < 4; ++ct) {             // 4 column tiles of 16
        v8f dacc = {0.f, 0.f, 0.f, 0.f, 0.f, 0.f, 0.f, 0.f};
        #pragma unroll
        for (int kc = 0; kc < 8; ++kc) {         // 8 K-chunks of 4
            v2f av; av.x = a0[kc]; av.y = a1[kc];
            v2f bv;
            bv.x = p0[kc][ct * 16];              // immediate-offset loads: off 0/64/128/192 B
            bv.y = p1[kc][ct * 16];
            dacc = __builtin_amdgcn_wmma_f32_16x16x4_f32(
                false, av, false, bv, (short)0, dacc, false, false);
        }
        r[ct] = dacc[0];                         // D row M=0: VGPR0, lanes 0..15 (N = lane)
    }

    // ---- Single guarded store block; output is write-once -> non-temporal ----
    float* orow = out + (size_t)qid * C_;
    if (lane < 16) {
        #pragma unroll
        for (int ct = 0; ct < 4; ++ct)
            __builtin_nontemporal_store(r[ct], orow + ct * 16 + lane);
    }
}

extern "C" void kernel_launch(void* const* d_in, const int* in_sizes, int n_in,
                              void* d_out, int out_size, void* d_ws, size_t ws_size,
                              hipStream_t stream) {
    const float*     query = (const float*)d_in[0];
    const float*     key   = (const float*)d_in[1];
    const float*     value = (const float*)d_in[2];
    const long long* pos   = (const long long*)d_in[3];
    float*           out   = (float*)d_out;

    const int Q = in_sizes[0] / C_;              // B*Lq*H = 65536
    const int blocks = (Q + WAVES - 1) / WAVES;  // 8192 blocks x 256 threads
    topk_attn_wmma_kernel<<<blocks, WAVES * 32, 0, stream>>>(query, key, value, pos, out, Q);
}